// hinge_mc_v2_64244120813574
// MI455X (gfx1250) — compile-verified
//
#include <hip/hip_runtime.h>

// CDNA5 / gfx1250: wave32. One wave per row of the [B, C] outputs matrix.
typedef float v2f __attribute__((ext_vector_type(2)));
typedef float v4f __attribute__((ext_vector_type(4)));
typedef float v8f __attribute__((ext_vector_type(8)));

#define WAVE 32
#define ROWS_PER_BLOCK 8   // 256 threads = 8 wave32s

// Wave-level exact-f32 sum of one value per lane using V_WMMA_F32_16X16X4_F32.
// A layout (ISA 7.12.2): lane L holds A[L%16][2*(L/16)+0] in comp x and
// A[L%16][2*(L/16)+1] in comp y. With B = all-ones, D[m][n] = sum_k A[m][k].
// Putting the partial p in both A components gives D[m][n] = 2*(p_m + p_{m+16}).
// C/D layout: lane L(<16) VGPR v holds M=v; lane L(>=16) holds M=v+8, so
// summing the 8 D components in-lane and adding lane^16 covers all 16 M once.
__device__ inline float wave_sum_wmma(float p) {
    v2f a;  a.x = p;   a.y = p;
    v2f b;  b.x = 1.f; b.y = 1.f;
    v8f c = {};
    v8f d = __builtin_amdgcn_wmma_f32_16x16x4_f32(
        /*neg_a=*/false, a, /*neg_b=*/false, b,
        /*c_mod=*/(short)0, c, /*reuse_a=*/false, /*reuse_b=*/false);
    float s = 0.f;
#pragma unroll
    for (int v = 0; v < 8; ++v) s += d[v];
    s += __shfl_xor(s, 16, WAVE);
    return 0.5f * s;
}

__global__ __launch_bounds__(256) void hinge_rows_kernel(
    const float* __restrict__ outputs, const int* __restrict__ labels,
    float* __restrict__ seg_sum, int* __restrict__ cnt,
    double* __restrict__ bin_acc, int B, int C)
{
    __shared__ double s_bin[ROWS_PER_BLOCK];
    const int waveId = (int)(threadIdx.x / WAVE);
    const int lane   = (int)(threadIdx.x % WAVE);
    const int row    = (int)blockIdx.x * ROWS_PER_BLOCK + waveId;

    if (row < B) {   // wave-uniform: B % ROWS_PER_BLOCK == 0 in practice
        const float* rp  = outputs + (size_t)row * (size_t)C;
        const int    lbl = labels[row];
        const float  own = rp[lbl];          // same addr for all lanes -> 1 request
        const float  co  = 1.f - own;        // pair-hinge offset

        float binP = 0.f, pairP = 0.f;

        // Hot loop: branch-free. The j == label column is folded in
        // unconditionally and corrected once per row after the reduction:
        //   bin : sum_j relu(1 + x_j)   - relu(1+own) + relu(1-own)
        //   pair: sum_j relu(1-own+x_j) - relu(1)  [ = 1 = MARGIN ]
        // float4 per lane -> 512B per wave per step; NT hint since the 262MB
        // stream exceeds the 192MB L2 and has no reuse.
        const int j4end = (C / (WAVE * 4)) * (WAVE * 4);
        for (int j = lane * 4; j < j4end; j += WAVE * 4) {
            v4f x = __builtin_nontemporal_load((const v4f*)(rp + j));
#pragma unroll
            for (int u = 0; u < 4; ++u) {
                const float xv = x[u];
                binP  += fmaxf(1.f + xv, 0.f);
                pairP += fmaxf(co  + xv, 0.f);
            }
        }
        // Tail (C % 128 columns)
        for (int j = j4end + lane; j < C; j += WAVE) {
            const float xv = rp[j];
            binP  += fmaxf(1.f + xv, 0.f);
            pairP += fmaxf(co  + xv, 0.f);
        }

        // Cross-lane reductions through the matrix pipe (EXEC all ones here).
        const float rowBin  = wave_sum_wmma(binP);
        const float rowPair = wave_sum_wmma(pairP);

        if (lane == 0) {
            // Per-row label-column corrections (exact terms, O(1) magnitude
            // against O(C) sums -> negligible rounding impact).
            const float rowBinC  = rowBin - fmaxf(1.f + own, 0.f)
                                          + fmaxf(1.f - own, 0.f);
            const float rowPairC = rowPair - 1.f;
            atomicAdd(&seg_sum[lbl], rowPairC);
            atomicAdd(&cnt[lbl], 1);
            s_bin[waveId] = (double)rowBinC;
        }
    } else {
        if (lane == 0) s_bin[waveId] = 0.0;
    }

    __syncthreads();
    if (threadIdx.x == 0) {
        double t = 0.0;
#pragma unroll
        for (int w = 0; w < ROWS_PER_BLOCK; ++w) t += s_bin[w];
        atomicAdd(bin_acc, t);   // global_atomic_add_f64
    }
}

__global__ __launch_bounds__(256) void hinge_finalize_kernel(
    const float* __restrict__ seg_sum, const int* __restrict__ cnt,
    const double* __restrict__ bin_acc, float* __restrict__ out, int B, int C)
{
    __shared__ double sh[256];
    double acc = 0.0;
    for (int c = (int)threadIdx.x; c < C; c += 256) {
        const int n = cnt[c];
        if (n > 0) acc += (double)seg_sum[c] / ((double)n * (double)(C - 1));
    }
    sh[threadIdx.x] = acc;
    __syncthreads();
    for (int s = 128; s > 0; s >>= 1) {
        if ((int)threadIdx.x < s) sh[threadIdx.x] += sh[threadIdx.x + s];
        __syncthreads();
    }
    if (threadIdx.x == 0) {
        const double bin = bin_acc[0] / ((double)B * (double)C);
        out[0] = (float)(bin + sh[0] / (double)C);
    }
}

extern "C" void kernel_launch(void* const* d_in, const int* in_sizes, int n_in,
                              void* d_out, int out_size, void* d_ws, size_t ws_size,
                              hipStream_t stream) {
    const float* outputs = (const float*)d_in[0];
    const int*   labels  = (const int*)d_in[1];
    const int B = in_sizes[1];
    const int C = in_sizes[0] / B;

    // Workspace layout: [0,8) double bin_acc | [8, 8+4C) float seg_sum | next 4C int cnt
    double* bin_acc = (double*)d_ws;
    float*  seg_sum = (float*)((char*)d_ws + 8);
    int*    cnt     = (int*)((char*)d_ws + 8 + sizeof(float) * (size_t)C);

    hipMemsetAsync(d_ws, 0, 8 + 8 * (size_t)C, stream);

    const int blocks = (B + ROWS_PER_BLOCK - 1) / ROWS_PER_BLOCK;
    hinge_rows_kernel<<<blocks, 256, 0, stream>>>(outputs, labels, seg_sum, cnt,
                                                  bin_acc, B, C);
    hinge_finalize_kernel<<<1, 256, 0, stream>>>(seg_sum, cnt, bin_acc,
                                                 (float*)d_out, B, C);
}